// PointnetSAModuleMSG_49606872269287
// MI455X (gfx1250) — compile-verified
//
#include <hip/hip_runtime.h>
#include <hip/hip_bf16.h>

typedef __attribute__((ext_vector_type(2))) float v2f;
typedef __attribute__((ext_vector_type(8))) float v8f;

#define B_   16
#define N_   2048
#define C_   128
#define NPT  512
#define CIN  131      // C + 3
#define CINP 132      // padded K
#define CMID 32
#define COUT 128

// ---------------------------------------------------------------------------
// Kernel 0: pad Wcr (131x128) -> wcrPad (132x128), last row zero
// ---------------------------------------------------------------------------
__global__ void pad_wcr_kernel(const float* __restrict__ Wcr, float* __restrict__ wcrPad) {
    int i = blockIdx.x * 256 + threadIdx.x;
    if (i < CINP * COUT)
        wcrPad[i] = (i < CIN * COUT) ? Wcr[i] : 0.0f;
}

// ---------------------------------------------------------------------------
// Kernel 1: furthest point sampling. One block per batch; xyz + dist in LDS.
// Writes fps indices to ws, new_xyz to ws and to head of d_out.
// ---------------------------------------------------------------------------
__global__ void fps_kernel(const float* __restrict__ xyz, int* __restrict__ fidx,
                           float* __restrict__ newxyz_ws, float* __restrict__ out_newxyz) {
    int b = blockIdx.x;
    __shared__ float sx[N_], sy[N_], sz[N_];
    __shared__ float sdist[N_];
    __shared__ float rval[256];
    __shared__ int   ridx[256];
    __shared__ int   s_last;

    const float* X = xyz + (size_t)b * N_ * 3;
    for (int i = threadIdx.x; i < N_; i += 256) {
        sx[i] = X[i * 3 + 0];
        sy[i] = X[i * 3 + 1];
        sz[i] = X[i * 3 + 2];
        sdist[i] = 1e10f;
    }
    if (threadIdx.x == 0) { s_last = 0; fidx[b * NPT + 0] = 0; }
    __syncthreads();

    for (int it = 1; it < NPT; ++it) {
        int last = s_last;
        float px = sx[last], py = sy[last], pz = sz[last];
        float best = -1.0f; int bi = 0;
        for (int i = threadIdx.x; i < N_; i += 256) {
            float dx = sx[i] - px, dy = sy[i] - py, dz = sz[i] - pz;
            float d = dx * dx + dy * dy + dz * dz;
            float m = fminf(sdist[i], d);
            sdist[i] = m;
            if (m > best) { best = m; bi = i; }
        }
        rval[threadIdx.x] = best; ridx[threadIdx.x] = bi;
        __syncthreads();
        for (int s = 128; s > 0; s >>= 1) {
            if ((int)threadIdx.x < s) {
                float ov = rval[threadIdx.x + s]; int oi = ridx[threadIdx.x + s];
                if (ov > rval[threadIdx.x] ||
                    (ov == rval[threadIdx.x] && oi < ridx[threadIdx.x])) {
                    rval[threadIdx.x] = ov; ridx[threadIdx.x] = oi;
                }
            }
            __syncthreads();
        }
        if (threadIdx.x == 0) { s_last = ridx[0]; fidx[b * NPT + it] = ridx[0]; }
        __syncthreads();
    }

    // emit new_xyz from LDS-cached coordinates
    for (int j = threadIdx.x; j < NPT; j += 256) {
        int id = fidx[b * NPT + j];
        float gx = sx[id], gy = sy[id], gz = sz[id];
        size_t o = (size_t)b * NPT * 3 + (size_t)j * 3;
        newxyz_ws[o + 0] = gx; newxyz_ws[o + 1] = gy; newxyz_ws[o + 2] = gz;
        out_newxyz[o + 0] = gx; out_newxyz[o + 1] = gy; out_newxyz[o + 2] = gz;
    }
}

// ---------------------------------------------------------------------------
// Kernel 2: per (center, batch) ball query + grouping + shared MLP + max-pool.
// Ball query uses a 2048-bit LDS mask + prefix sums so sample order equals the
// reference's first-ns-by-index semantics. Writes pooled[(b*512+p)*132 + k],
// with pad element k=131 set to 0 for the WMMA K-padding.
// ---------------------------------------------------------------------------
__global__ void group_mlp_kernel(const float* __restrict__ xyz,
                                 const float* __restrict__ features,
                                 const float* __restrict__ newxyz,
                                 const float* __restrict__ W1, const float* __restrict__ b1,
                                 const float* __restrict__ W2, const float* __restrict__ b2,
                                 float* __restrict__ pooled,
                                 float r2, int ns) {
    int p = blockIdx.x;   // center index 0..511
    int b = blockIdx.y;   // batch
    int tid = threadIdx.x;

    __shared__ unsigned smask[64];
    __shared__ int      scnt[64];
    __shared__ int      sidx[64];      // ns <= 48
    __shared__ int      s_total;
    __shared__ float    h10s[48][10];
    __shared__ float    t1[48][CMID];
    __shared__ float    sW1[10 * CMID];
    __shared__ float    sb1[CMID];
    __shared__ float    sW2[CMID * CIN];
    __shared__ float    sb2[CIN];

    // stage tiny weights into LDS
    for (int i = tid; i < 10 * CMID; i += 256) sW1[i] = W1[i];
    for (int i = tid; i < CMID;      i += 256) sb1[i] = b1[i];
    for (int i = tid; i < CMID * CIN; i += 256) sW2[i] = W2[i];
    for (int i = tid; i < CIN;       i += 256) sb2[i] = b2[i];

    size_t cOff = ((size_t)b * NPT + p) * 3;
    float cx = newxyz[cOff + 0], cy = newxyz[cOff + 1], cz = newxyz[cOff + 2];

    // build in-ball bitmask: thread t owns points [t*32, t*32+32)
    const float* X = xyz + (size_t)b * N_ * 3;
    if (tid < 64) {
        unsigned w = 0u;
        int base = tid * 32;
        for (int j = 0; j < 32; ++j) {
            int i = base + j;
            float dx = X[i * 3 + 0] - cx;
            float dy = X[i * 3 + 1] - cy;
            float dz = X[i * 3 + 2] - cz;
            float d2 = dx * dx + dy * dy + dz * dz;
            w |= (d2 < r2 ? 1u : 0u) << j;
        }
        smask[tid] = w;
    }
    __syncthreads();

    // serial 64-word prefix (cheap) by thread 0
    if (tid == 0) {
        int run = 0;
        for (int w = 0; w < 64; ++w) { scnt[w] = run; run += __popc(smask[w]); }
        s_total = run;
        sidx[0] = 0; // default if empty ball (never happens: center is in ball)
    }
    __syncthreads();

    // emit the first ns indices in index order
    if (tid < 64) {
        unsigned w = smask[tid];
        int rank = scnt[tid];
        while (w) {
            int bit = __ffs(w) - 1;
            w &= (w - 1u);
            if (rank < ns) sidx[rank] = tid * 32 + bit;
            ++rank;
        }
    }
    __syncthreads();

    if (tid == 0) {
        int tot = s_total < ns ? s_total : ns;
        if (tot == 0) tot = 1;
        int first = sidx[0];
        for (int i = tot; i < ns; ++i) sidx[i] = first;
    }
    __syncthreads();

    // h10 = [dist, center(3), g_xyz(3), rel(3)] per sample
    if (tid < ns) {
        int id = sidx[tid];
        float gx = X[id * 3 + 0], gy = X[id * 3 + 1], gz = X[id * 3 + 2];
        float rx = gx - cx, ry = gy - cy, rz = gz - cz;
        float dist = sqrtf(rx * rx + ry * ry + rz * rz);
        h10s[tid][0] = dist;
        h10s[tid][1] = cx; h10s[tid][2] = cy; h10s[tid][3] = cz;
        h10s[tid][4] = gx; h10s[tid][5] = gy; h10s[tid][6] = gz;
        h10s[tid][7] = rx; h10s[tid][8] = ry; h10s[tid][9] = rz;
    }
    __syncthreads();

    // t1[s][j] = relu(h10[s] @ W1 + b1)
    for (int q = tid; q < ns * CMID; q += 256) {
        int s = q >> 5, j = q & 31;
        float acc = sb1[j];
        #pragma unroll
        for (int k = 0; k < 10; ++k) acc += h10s[s][k] * sW1[k * CMID + j];
        t1[s][j] = acc > 0.0f ? acc : 0.0f;
    }
    __syncthreads();

    // pooled[k] = max_s relu( (t1[s] @ W2 + b2)[k] * x[s][k] )
    const float* F = features + (size_t)b * C_ * N_;
    if (tid < CIN) {
        int k = tid;
        float m = 0.0f;  // relu outputs are >= 0
        for (int s = 0; s < ns; ++s) {
            float h = sb2[k];
            #pragma unroll
            for (int j = 0; j < CMID; ++j) h += t1[s][j] * sW2[j * CIN + k];
            float x = (k < 3) ? h10s[s][7 + k] : F[(size_t)(k - 3) * N_ + sidx[s]];
            float v = h * x;
            v = v > 0.0f ? v : 0.0f;
            m = fmaxf(m, v);
        }
        pooled[((size_t)b * NPT + p) * CINP + k] = m;
    } else if (tid == CIN) {
        pooled[((size_t)b * NPT + p) * CINP + CIN] = 0.0f;   // K-pad lane
    }
}

// ---------------------------------------------------------------------------
// Kernel 3: out = relu(pooled @ WcrPad + bcr) via V_WMMA_F32_16X16X4_F32.
// Grid: (512 M-tiles, 3 radii). Block: 256 = 8 waves; wave w owns N-tile w
// (8*16 = 128 = full N). A 16x132 tile staged in LDS, K stepped by 4.
// Output scattered into (B, 384, 512)-transposed layout.
// ---------------------------------------------------------------------------
__global__ void gemm_wmma_kernel(const float* __restrict__ pooledAll,
                                 const float* __restrict__ wcrPad,
                                 const float* __restrict__ bcr,
                                 float* __restrict__ outFeat) {
    int mtile = blockIdx.x;        // 0..511 over M = B*NPT = 8192
    int r     = blockIdx.y;        // radius index 0..2
    const float* pooled = pooledAll + (size_t)r * (B_ * NPT) * CINP;

    __shared__ float At[16 * CINP];
    int tid = threadIdx.x;
    const float* Ag = pooled + (size_t)mtile * 16 * CINP;
    for (int i = tid; i < 16 * CINP; i += 256) At[i] = Ag[i];
    __syncthreads();

    int lane = tid & 31;
    int wv   = tid >> 5;           // N-tile index 0..7
    int l16  = lane & 15;
    int hiK  = (lane >> 4) << 1;   // 0 for lanes 0-15, 2 for lanes 16-31
    int col  = wv * 16 + l16;      // output channel within [0,128)

    v8f acc = {};
    #pragma unroll 4
    for (int k = 0; k < CINP; k += 4) {
        // A fragment (16x4, M x K): lane<16 carries K=k,k+1; lane>=16 K=k+2,k+3
        v2f Af;
        Af.x = At[l16 * CINP + k + hiK + 0];
        Af.y = At[l16 * CINP + k + hiK + 1];
        // B fragment (4x16, K x N), mirrored striping
        v2f Bf;
        Bf.x = wcrPad[(k + hiK + 0) * COUT + col];
        Bf.y = wcrPad[(k + hiK + 1) * COUT + col];
        acc = __builtin_amdgcn_wmma_f32_16x16x4_f32(
            /*neg_a=*/false, Af, /*neg_b=*/false, Bf,
            /*c_mod=*/(short)0, acc, /*reuse_a=*/false, /*reuse_b=*/false);
    }

    float bias = bcr[col];
    int rowAdd = (lane >> 4) << 3; // lanes>=16 hold M rows 8..15 of the tile
    #pragma unroll
    for (int v = 0; v < 8; ++v) {
        int m = mtile * 16 + v + rowAdd;   // global row = b*512 + p
        int b = m >> 9;
        int p = m & 511;
        float val = acc[v] + bias;
        val = val > 0.0f ? val : 0.0f;
        // new_features layout (B, 384, 512); radius r occupies channels [r*128, r*128+128)
        outFeat[(size_t)b * (3 * COUT) * NPT + (size_t)(r * COUT + col) * NPT + p] = val;
    }
}

// ---------------------------------------------------------------------------
extern "C" void kernel_launch(void* const* d_in, const int* in_sizes, int n_in,
                              void* d_out, int out_size, void* d_ws, size_t ws_size,
                              hipStream_t stream) {
    const float* xyz      = (const float*)d_in[0];  // (B, N, 3)
    const float* features = (const float*)d_in[1];  // (B, C, N)
    const float* W1       = (const float*)d_in[2];  // (10, 32)
    const float* b1       = (const float*)d_in[3];  // (32,)
    const float* W2       = (const float*)d_in[4];  // (32, 131)
    const float* b2       = (const float*)d_in[5];  // (131,)
    const float* Wcr      = (const float*)d_in[6];  // (131, 128)
    const float* bcr      = (const float*)d_in[7];  // (128,)

    float* out      = (float*)d_out;
    float* out_nxyz = out;                            // (B, 512, 3)
    float* out_feat = out + (size_t)B_ * NPT * 3;     // (B, 384, 512)

    // workspace carve-out (all 256B aligned)
    char* ws = (char*)d_ws;
    size_t o = 0;
    int*   fidx    = (int*)(ws + o);   o += (size_t)B_ * NPT * sizeof(int);        o = (o + 255) & ~(size_t)255;
    float* nxyz_ws = (float*)(ws + o); o += (size_t)B_ * NPT * 3 * sizeof(float);  o = (o + 255) & ~(size_t)255;
    float* wcrPad  = (float*)(ws + o); o += (size_t)CINP * COUT * sizeof(float);   o = (o + 255) & ~(size_t)255;
    float* pooled  = (float*)(ws + o); // 3 * B*NPT * CINP floats (~13 MB)

    // 0) pad Wcr with a zero K-row
    pad_wcr_kernel<<<(CINP * COUT + 255) / 256, 256, 0, stream>>>(Wcr, wcrPad);

    // 1) furthest point sampling
    fps_kernel<<<B_, 256, 0, stream>>>(xyz, fidx, nxyz_ws, out_nxyz);

    // 2) ball query + grouping + MLP + max-pool, per radius
    const float radii[3]   = {0.1f, 0.2f, 0.4f};
    const int   nsamp[3]   = {16, 32, 48};
    for (int r = 0; r < 3; ++r) {
        float* pooled_r = pooled + (size_t)r * (B_ * NPT) * CINP;
        group_mlp_kernel<<<dim3(NPT, B_), 256, 0, stream>>>(
            xyz, features, nxyz_ws, W1, b1, W2, b2,
            pooled_r, radii[r] * radii[r], nsamp[r]);
    }

    // 3) fused GEMM + bias + relu + transpose via f32 WMMA
    gemm_wmma_kernel<<<dim3((B_ * NPT) / 16, 3), 256, 0, stream>>>(
        pooled, wcrPad, bcr, out_feat);
}